// CBERTLinear_11029476016379
// MI455X (gfx1250) — compile-verified
//
#include <hip/hip_runtime.h>
#include <cstdint>
#include <cstddef>

// ---------------------------------------------------------------------------
// CDNA5 / gfx1250 implementation of the ragged-candidate sense-classification
// loss.  One 64-thread block (2 wave32) per sample:
//   phase 0: reps[b] -> LDS, sense ids -> LDS
//   phase 1: gather valid candidate rows -> LDS (global_load_async_to_lds_b128)
//   phase 2: logits via V_WMMA_F32_16X16X4_F32 with 4 independent accumulators
//            (B-matrix = reps broadcast across all 16 N columns, so every
//            column equals the exact fp32 logit)
//   phase 3: wave32 softmax / NLL / first-argmax
// Second kernel: deterministic mean reduction of the NLL into d_out[0].
// ---------------------------------------------------------------------------

typedef __attribute__((ext_vector_type(2))) float v2f;
typedef __attribute__((ext_vector_type(4))) float v4f;
typedef __attribute__((ext_vector_type(8))) float v8f;
typedef __attribute__((ext_vector_type(4))) int   v4i;

#ifndef __has_builtin
#define __has_builtin(x) 0
#endif

#if __has_builtin(__builtin_amdgcn_global_load_async_to_lds_b128)
#define HAVE_ASYNC_LDS 1
#else
#define HAVE_ASYNC_LDS 0
#endif

#if HAVE_ASYNC_LDS
static __device__ __forceinline__ void wait_async_zero() {
#if __has_builtin(__builtin_amdgcn_s_wait_asynccnt)
  __builtin_amdgcn_s_wait_asynccnt(0);
#else
  asm volatile("s_wait_asynccnt 0" ::: "memory");
#endif
}
#endif

static __device__ __forceinline__ v8f wmma_step(const float* wp, const float* rp,
                                                v8f acc) {
  v2f a  = *(const v2f*)wp;   // A fragment (candidate weights)
  v2f bb = *(const v2f*)rp;   // B fragment (reps, broadcast over N)
  return __builtin_amdgcn_wmma_f32_16x16x4_f32(
      false, a, false, bb, (short)0, acc, false, false);
}

__global__ __launch_bounds__(64) void sense_ce_kernel(
    const float* __restrict__ reps,        // [B, 768]
    const float* __restrict__ weight,      // [N_SENSES, 768]
    const float* __restrict__ bias,        // [N_SENSES]
    const int*   __restrict__ sense_ids,   // [B, 32]
    const int*   __restrict__ target_ids,  // [B]
    float*       __restrict__ out,         // [1 + B]; out[1+b] = correct
    float*       __restrict__ nll)         // [B] scratch
{
  constexpr int D       = 768;
  constexpr int K       = 32;
  constexpr int CHUNK   = 384;           // d-values staged per pass
  constexpr int WSTRIDE = CHUNK + 4;     // 388: 4r-mod-64 banking -> no conflicts
  constexpr int QPC     = CHUNK / 4;     // float4 segments per row per chunk

  __shared__ float lds_w[K * WSTRIDE];   // 49,664 B
  __shared__ float lds_reps[D];          //  3,072 B
  __shared__ float lds_logits[K];
  __shared__ int   lds_ids[K];

  const int b    = blockIdx.x;
  const int tid  = threadIdx.x;
  const int lane = tid & 31;
  const int wv   = tid >> 5;             // wave -> candidate tile (0: k<16, 1: k>=16)

  if (tid < K) lds_ids[tid] = sense_ids[b * K + tid];
  for (int i = tid; i < D / 4; i += 64) {
    *(v4f*)&lds_reps[i * 4] = *(const v4f*)(reps + (size_t)b * D + i * 4);
  }
  __syncthreads();

  // WMMA fragment ownership (ISA 7.12.2, fp32 16x16x4):
  //   A: lane L holds row M = L%16, K-pair koff = 2*(L/16)  -> 2 VGPRs
  //   B: lane half h supplies rows K = 2h, 2h+1 (broadcast over N columns)
  const int m    = lane & 15;
  const int koff = (lane >> 4) << 1;
  const float* wrow = &lds_w[(wv * 16 + m) * WSTRIDE + koff];

  v8f acc0 = {}, acc1 = {}, acc2 = {}, acc3 = {};

  for (int c = 0; c < D / CHUNK; ++c) {
    // ---- gather valid candidate rows (one chunk) into LDS, coalesced B128 ----
    for (int idx = tid; idx < K * QPC; idx += 64) {
      const int row = idx / QPC;
      const int q   = idx - row * QPC;
      const int id  = lds_ids[row];
      if (id >= 0) {
        const float* g = weight + (size_t)id * D + c * CHUNK + q * 4;
        float*       l = &lds_w[row * WSTRIDE + q * 4];
#if HAVE_ASYNC_LDS
        __builtin_amdgcn_global_load_async_to_lds_b128((v4i*)g, (v4i*)l, 0, 0);
#else
        *(v4f*)l = *(const v4f*)g;
#endif
      }
    }
#if HAVE_ASYNC_LDS
    wait_async_zero();
#endif
    __syncthreads();

    // ---- WMMA K-steps over this chunk: 4 independent accumulator chains ----
    const float* rp = &lds_reps[c * CHUNK + koff];
#pragma unroll 4
    for (int d0 = 0; d0 < CHUNK; d0 += 16) {
      acc0 = wmma_step(wrow + d0,      rp + d0,      acc0);
      acc1 = wmma_step(wrow + d0 + 4,  rp + d0 + 4,  acc1);
      acc2 = wmma_step(wrow + d0 + 8,  rp + d0 + 8,  acc2);
      acc3 = wmma_step(wrow + d0 + 12, rp + d0 + 12, acc3);
    }
    __syncthreads();
  }

  const v8f acc = (acc0 + acc1) + (acc2 + acc3);

  // D layout: VGPR j holds rows {j, j+8} (lane halves); every column identical.
  if ((lane & 15) == 0) {
    const int base = wv * 16 + (lane >> 4) * 8;
#pragma unroll
    for (int j = 0; j < 8; ++j) lds_logits[base + j] = acc[j];
  }
  __syncthreads();

  // ---- wave0: bias + mask, softmax-CE, first-argmax ----
  if (tid < K) {
    const int id = lds_ids[tid];
    float l = (id >= 0) ? (lds_logits[tid] + bias[id]) : -1e30f;

    float mx = l;
    for (int o = 16; o > 0; o >>= 1) mx = fmaxf(mx, __shfl_xor(mx, o, 32));
    float s = expf(l - mx);
    for (int o = 16; o > 0; o >>= 1) s += __shfl_xor(s, o, 32);
    const float logZ = mx + logf(s);

    const int tgt = target_ids[b];
    if (tid == tgt) nll[b] = logZ - l;   // -log_softmax at the target

    float bv = l;
    int   bi = tid;
    for (int o = 16; o > 0; o >>= 1) {
      const float ov = __shfl_xor(bv, o, 32);
      const int   oi = __shfl_xor(bi, o, 32);
      if (ov > bv || (ov == bv && oi < bi)) { bv = ov; bi = oi; }
    }
    if (tid == 0) out[1 + b] = (bi == tgt) ? 1.0f : 0.0f;
  }
}

__global__ __launch_bounds__(256) void loss_reduce_kernel(
    const float* __restrict__ nll, float* __restrict__ out, int B)
{
  __shared__ float sm[256];
  const int t = threadIdx.x;
  float v = 0.0f;
  for (int i = t; i < B; i += 256) v += nll[i];
  sm[t] = v;
  __syncthreads();
  for (int s = 128; s > 0; s >>= 1) {
    if (t < s) sm[t] += sm[t + s];
    __syncthreads();
  }
  if (t == 0) out[0] = sm[0] / (float)B;
}

extern "C" void kernel_launch(void* const* d_in, const int* in_sizes, int n_in,
                              void* d_out, int out_size, void* d_ws, size_t ws_size,
                              hipStream_t stream) {
  const float* reps       = (const float*)d_in[0];
  const float* weight     = (const float*)d_in[1];
  const float* bias       = (const float*)d_in[2];
  const int*   sense_ids  = (const int*)d_in[3];
  const int*   target_ids = (const int*)d_in[4];
  float* out = (float*)d_out;
  float* nll = (float*)d_ws;
  const int B = in_sizes[4];   // target_ids has one entry per sample

  sense_ce_kernel<<<B, 64, 0, stream>>>(reps, weight, bias, sense_ids,
                                        target_ids, out, nll);
  loss_reduce_kernel<<<1, 256, 0, stream>>>(nll, out, B);
}